// ViTE_58342835749147
// MI455X (gfx1250) — compile-verified
//
#include <hip/hip_runtime.h>
#include <math.h>

// Problem constants (match reference)
#define B_   32
#define N_   512
#define V_   5
#define D_   256
#define H_   4
#define BN_  (B_ * N_)          // 16384 real nodes
#define T_   (BN_ + B_ * V_)    // 16544 total nodes (multiple of 16)
#define HD_  (H_ * D_)          // 1024
#define EPS_ 1e-5f

typedef float v2f __attribute__((ext_vector_type(2)));
typedef float v8f __attribute__((ext_vector_type(8)));

__device__ __forceinline__ float gelu_exact(float x) {
    return 0.5f * x * (1.0f + erff(x * 0.70710678118654752f));
}

// ---------------------------------------------------------------------------
// K1: all_nodes = concat(x, tile(virtual_nodes, B))
// ---------------------------------------------------------------------------
__global__ void k_build_nodes(const float* __restrict__ x,
                              const float* __restrict__ vn,
                              float* __restrict__ nodes) {
    size_t i = (size_t)blockIdx.x * blockDim.x + threadIdx.x;
    if (i < (size_t)BN_ * D_) {
        nodes[i] = x[i];
    } else if (i < (size_t)T_ * D_) {
        size_t j = i - (size_t)BN_ * D_;
        nodes[i] = vn[j % (V_ * D_)];   // (b*V+v)*D+d -> v*D+d
    }
}

// ---------------------------------------------------------------------------
// K2: fp32 WMMA GEMM  C[M,N] = A[M,K] * Bm[K,N]   (row-major, all dims %16==0)
// block = 128 threads (4 waves). Each wave computes a 16x64 tile via 4 WMMAs
// per K-step; A fragment loaded once per step and reused for all 4 N-subtiles.
// V_WMMA_F32_16X16X4_F32 fragment layout (ISA 7.12.2):
//   A 16x4:  lanes 0-15 -> row M=lane, {K=k0,k0+1}; lanes 16-31 -> {k0+2,k0+3}
//   B 4x16:  lanes 0-15 -> col N=lane, {K=k0,k0+1}; lanes 16-31 -> {k0+2,k0+3}
//   C/D:     VGPR j -> row m0 + 8*(lane>=16) + j, col n0 + (lane&15)
// ---------------------------------------------------------------------------
template <int KDIM, int NDIM>
__global__ void k_gemm_f32_wmma(const float* __restrict__ A,
                                const float* __restrict__ Bm,
                                float* __restrict__ C) {
    const int wave = threadIdx.x >> 5;
    const int lane = threadIdx.x & 31;
    const int half = lane >> 4;
    const int lm   = lane & 15;
    const int m0   = blockIdx.x * 16;
    const int n0   = blockIdx.y * 256 + wave * 64;
    if (n0 >= NDIM) return;

    v8f c0 = {}; v8f c1 = {}; v8f c2 = {}; v8f c3 = {};
    const float* arow = A + (size_t)(m0 + lm) * KDIM;

    for (int k0 = 0; k0 < KDIM; k0 += 4) {
        const int kb = k0 + half * 2;
        v2f a;
        a[0] = arow[kb];
        a[1] = arow[kb + 1];
        const float* b0p = Bm + (size_t)kb * NDIM + n0 + lm;
        const float* b1p = b0p + NDIM;
        v2f b;
        b[0] = b0p[0];  b[1] = b1p[0];
        c0 = __builtin_amdgcn_wmma_f32_16x16x4_f32(false, a, false, b, (short)0, c0, false, false);
        b[0] = b0p[16]; b[1] = b1p[16];
        c1 = __builtin_amdgcn_wmma_f32_16x16x4_f32(false, a, false, b, (short)0, c1, false, false);
        b[0] = b0p[32]; b[1] = b1p[32];
        c2 = __builtin_amdgcn_wmma_f32_16x16x4_f32(false, a, false, b, (short)0, c2, false, false);
        b[0] = b0p[48]; b[1] = b1p[48];
        c3 = __builtin_amdgcn_wmma_f32_16x16x4_f32(false, a, false, b, (short)0, c3, false, false);
    }

    float* crow = C + n0 + lm;
#pragma unroll
    for (int j = 0; j < 8; ++j) {
        const size_t row = (size_t)(m0 + half * 8 + j) * NDIM;
        crow[row + 0]  = c0[j];
        crow[row + 16] = c1[j];
        crow[row + 32] = c2[j];
        crow[row + 48] = c3[j];
    }
}

// ---------------------------------------------------------------------------
// K3: a_src[t,h] = <h[t,h,:], att_src[h,:]>, same for a_dst. One block per t.
// ---------------------------------------------------------------------------
__global__ void k_scores(const float* __restrict__ h,
                         const float* __restrict__ att_src,
                         const float* __restrict__ att_dst,
                         float* __restrict__ a_src,
                         float* __restrict__ a_dst) {
    __shared__ float red[D_];
    const int t = blockIdx.x;
    const int d = threadIdx.x;
    const float* hr = h + (size_t)t * HD_;
#pragma unroll
    for (int hd = 0; hd < H_; ++hd) {
        const float hv = hr[hd * D_ + d];
        red[d] = hv * att_src[hd * D_ + d];
        __syncthreads();
        for (int s = D_ / 2; s > 0; s >>= 1) {
            if (d < s) red[d] += red[d + s];
            __syncthreads();
        }
        if (d == 0) a_src[t * H_ + hd] = red[0];
        __syncthreads();
        red[d] = hv * att_dst[hd * D_ + d];
        __syncthreads();
        for (int s = D_ / 2; s > 0; s >>= 1) {
            if (d < s) red[d] += red[d + s];
            __syncthreads();
        }
        if (d == 0) a_dst[t * H_ + hd] = red[0];
        __syncthreads();
    }
}

// ---------------------------------------------------------------------------
// K4: GAT softmax-aggregation — only virtual destinations need it.
// Block (uidx, head): softmax over 512 in-batch reals + self loop,
// then each thread owns one channel d and accumulates.
// ---------------------------------------------------------------------------
__global__ void k_gat_virtual(const float* __restrict__ h,
                              const float* __restrict__ a_src,
                              const float* __restrict__ a_dst,
                              float* __restrict__ gat_agg) {
    __shared__ float w[N_ + 1];
    __shared__ float red[D_];
    const int uidx = blockIdx.x;          // 0..B*V-1
    const int head = blockIdx.y;          // 0..H-1
    const int b    = uidx / V_;
    const int u    = BN_ + uidx;
    const int tid  = threadIdx.x;

    const float adst = a_dst[u * H_ + head];
    const int r0 = b * N_ + tid * 2;
    float al0 = a_src[(size_t)r0 * H_ + head] + adst;
    al0 = (al0 > 0.0f) ? al0 : 0.2f * al0;
    float al1 = a_src[(size_t)(r0 + 1) * H_ + head] + adst;
    al1 = (al1 > 0.0f) ? al1 : 0.2f * al1;
    float aself = a_src[(size_t)u * H_ + head] + adst;
    aself = (aself > 0.0f) ? aself : 0.2f * aself;

    float lmax = fmaxf(al0, al1);
    if (tid == 0) lmax = fmaxf(lmax, aself);
    red[tid] = lmax;
    __syncthreads();
    for (int s = D_ / 2; s > 0; s >>= 1) {
        if (tid < s) red[tid] = fmaxf(red[tid], red[tid + s]);
        __syncthreads();
    }
    const float m = red[0];
    __syncthreads();

    const float e0 = expf(al0 - m);
    const float e1 = expf(al1 - m);
    w[tid * 2]     = e0;
    w[tid * 2 + 1] = e1;
    float es = 0.0f;
    if (tid == 0) { es = expf(aself - m); w[N_] = es; }
    red[tid] = e0 + e1 + es;
    __syncthreads();
    for (int s = D_ / 2; s > 0; s >>= 1) {
        if (tid < s) red[tid] += red[tid + s];
        __syncthreads();
    }
    const float inv = 1.0f / red[0];
    __syncthreads();

    const float* hb = h + (size_t)(b * N_) * HD_ + head * D_ + tid;
    float acc = 0.0f;
    for (int r = 0; r < N_; ++r)
        acc += w[r] * hb[(size_t)r * HD_];
    acc += w[N_] * h[(size_t)u * HD_ + head * D_ + tid];
    gat_agg[((size_t)uidx * H_ + head) * D_ + tid] = acc * inv;
}

// ---------------------------------------------------------------------------
// K5: gat_out = mean over heads (+bias) -> GELU -> LayerNorm. One block per t.
// Real nodes: mean of h[t]; virtual: mean of gat_agg.
// ---------------------------------------------------------------------------
__global__ void k_gat_finish(const float* __restrict__ h,
                             const float* __restrict__ gat_agg,
                             const float* __restrict__ bias,
                             const float* __restrict__ gg,
                             const float* __restrict__ gb,
                             float* __restrict__ out) {
    __shared__ float red[D_];
    const int t = blockIdx.x;
    const int d = threadIdx.x;
    float v;
    if (t < BN_) {
        const float* p = h + (size_t)t * HD_ + d;
        v = 0.25f * (p[0] + p[D_] + p[2 * D_] + p[3 * D_]);
    } else {
        const float* p = gat_agg + (size_t)(t - BN_) * HD_ + d;
        v = 0.25f * (p[0] + p[D_] + p[2 * D_] + p[3 * D_]);
    }
    v = gelu_exact(v + bias[d]);

    red[d] = v;
    __syncthreads();
    for (int s = D_ / 2; s > 0; s >>= 1) {
        if (d < s) red[d] += red[d + s];
        __syncthreads();
    }
    const float mu = red[0] * (1.0f / D_);
    __syncthreads();
    const float c = v - mu;
    red[d] = c * c;
    __syncthreads();
    for (int s = D_ / 2; s > 0; s >>= 1) {
        if (d < s) red[d] += red[d + s];
        __syncthreads();
    }
    const float var = red[0] * (1.0f / D_);
    out[(size_t)t * D_ + d] = c * rsqrtf(var + EPS_) * gg[d] + gb[d];
}

// ---------------------------------------------------------------------------
// K7: GCN combine (sym-norm: deg(real)=6, deg(virt)=1) -> GELU -> LayerNorm.
// Only real rows are emitted. One block per real node.
// ---------------------------------------------------------------------------
__global__ void k_gcn_finish(const float* __restrict__ xw,
                             const float* __restrict__ bias,
                             const float* __restrict__ gg,
                             const float* __restrict__ gb,
                             float* __restrict__ out) {
    __shared__ float red[D_];
    const int t = blockIdx.x;
    const int d = threadIdx.x;
    const int b = t / N_;
    const float* xv = xw + (size_t)(BN_ + b * V_) * D_ + d;
    const float s5 = xv[0] + xv[D_] + xv[2 * D_] + xv[3 * D_] + xv[4 * D_];
    const float inv_sqrt6 = 0.40824829046386301637f;
    float v = s5 * inv_sqrt6 + xw[(size_t)t * D_ + d] * (1.0f / 6.0f) + bias[d];
    v = gelu_exact(v);

    red[d] = v;
    __syncthreads();
    for (int s = D_ / 2; s > 0; s >>= 1) {
        if (d < s) red[d] += red[d + s];
        __syncthreads();
    }
    const float mu = red[0] * (1.0f / D_);
    __syncthreads();
    const float c = v - mu;
    red[d] = c * c;
    __syncthreads();
    for (int s = D_ / 2; s > 0; s >>= 1) {
        if (d < s) red[d] += red[d + s];
        __syncthreads();
    }
    const float var = red[0] * (1.0f / D_);
    out[(size_t)t * D_ + d] = c * rsqrtf(var + EPS_) * gg[d] + gb[d];
}

// ---------------------------------------------------------------------------
extern "C" void kernel_launch(void* const* d_in, const int* in_sizes, int n_in,
                              void* d_out, int out_size, void* d_ws, size_t ws_size,
                              hipStream_t stream) {
    (void)in_sizes; (void)n_in; (void)out_size; (void)ws_size;

    const float* x       = (const float*)d_in[0];
    /* d_in[1] = edge_index (unused by the reference computation) */
    const float* vn      = (const float*)d_in[2];
    const float* gat_W   = (const float*)d_in[3];   // [256, 1024]
    const float* att_src = (const float*)d_in[4];   // [4, 256]
    const float* att_dst = (const float*)d_in[5];   // [4, 256]
    const float* gat_b   = (const float*)d_in[6];   // [256]
    const float* gcn_W   = (const float*)d_in[7];   // [256, 256]
    const float* gcn_b   = (const float*)d_in[8];
    const float* n1g     = (const float*)d_in[9];
    const float* n1b     = (const float*)d_in[10];
    const float* n2g     = (const float*)d_in[11];
    const float* n2b     = (const float*)d_in[12];
    float* out = (float*)d_out;

    // Workspace layout (floats)
    float* ws = (float*)d_ws;
    size_t off = 0;
    float* nodes = ws + off; off += (size_t)T_ * D_;        // all_nodes
    float* h     = ws + off; off += (size_t)T_ * HD_;       // GAT hidden
    float* asrc  = ws + off; off += (size_t)T_ * H_;
    float* adst  = ws + off; off += (size_t)T_ * H_;
    float* gagg  = ws + off; off += (size_t)B_ * V_ * HD_;  // virtual GAT agg
    float* gat1  = ws + off; off += (size_t)T_ * D_;        // post-LN1
    float* xw    = ws + off; off += (size_t)T_ * D_;        // GCN xw

    // K1: assemble node matrix
    {
        const size_t n = (size_t)T_ * D_;
        k_build_nodes<<<(unsigned)((n + 255) / 256), 256, 0, stream>>>(x, vn, nodes);
    }
    // K2: h = nodes @ gat_W  (16544 x 1024, K=256) via fp32 WMMA
    {
        dim3 grid(T_ / 16, HD_ / 256);  // (1034, 4)
        k_gemm_f32_wmma<D_, HD_><<<grid, 128, 0, stream>>>(nodes, gat_W, h);
    }
    // K3: attention scores
    k_scores<<<T_, D_, 0, stream>>>(h, att_src, att_dst, asrc, adst);
    // K4: softmax-aggregate onto virtual nodes
    {
        dim3 grid(B_ * V_, H_);         // (160, 4)
        k_gat_virtual<<<grid, D_, 0, stream>>>(h, asrc, adst, gagg);
    }
    // K5: head-mean + bias + GELU + LayerNorm1
    k_gat_finish<<<T_, D_, 0, stream>>>(h, gagg, gat_b, n1g, n1b, gat1);
    // K6: xw = gat1 @ gcn_W  (16544 x 256, K=256) via fp32 WMMA
    {
        dim3 grid(T_ / 16, 1);
        k_gemm_f32_wmma<D_, D_><<<grid, 128, 0, stream>>>(gat1, gcn_W, xw);
    }
    // K7: GCN sym-norm combine + bias + GELU + LayerNorm2 -> real rows only
    k_gcn_finish<<<BN_, D_, 0, stream>>>(xw, gcn_b, n2g, n2b, out);
}